// GraphNeuralRecommender_26826365731399
// MI455X (gfx1250) — compile-verified
//
#include <hip/hip_runtime.h>
#include <hip/hip_bf16.h>
#include <math.h>

#define NUM_USERS 60000
#define NUM_ITEMS 40000
#define NUM_NODES (NUM_USERS + NUM_ITEMS)
#define EMB 64
#define HID 32
#define N_EDGES 1600000
#define BATCH 16384

typedef __attribute__((ext_vector_type(2))) float v2f;
typedef __attribute__((ext_vector_type(8))) float v8f;

// ---------------------------------------------------------------------------
// Utility kernels
// ---------------------------------------------------------------------------
__global__ void zero_kernel(float* __restrict__ p, int n) {
    int i = blockIdx.x * blockDim.x + threadIdx.x;
    if (i < n) p[i] = 0.0f;
}

__global__ void relu_kernel(float* __restrict__ p, int n) {
    int i = blockIdx.x * blockDim.x + threadIdx.x;
    if (i < n) p[i] = fmaxf(p[i], 0.0f);
}

// ---------------------------------------------------------------------------
// WMMA GEMM:  out[node, j] = sum_k act(in[node, k]) * W[j, k] + bias[j]
//   - per-wave 16xNOUT tile via V_WMMA_F32_16X16X4_F32 (exact fp32)
//   - 8 waves / block -> 128 nodes per block
//   - A tile staged in LDS (padded rows, conflict-free fragment reads)
//   - CONCAT: row < split reads inA, else inB (virtual concat of embeddings)
// ---------------------------------------------------------------------------
template <int K, int NOUT, bool RELU_IN, bool CONCAT>
__global__ __launch_bounds__(256) void gc_gemm(
    const float* __restrict__ inA, const float* __restrict__ inB, int split,
    const float* __restrict__ W, const float* __restrict__ bias,
    float* __restrict__ out, int nodes)
{
    constexpr int WAVES = 8;
    constexpr int MT    = 16;
    constexpr int NT    = NOUT / 16;
    constexpr int KP    = K + 4;          // padded row (floats), keeps 16B align
    constexpr int F4ROW = K / 4;

    __shared__ float sA[WAVES][MT][KP];
    __shared__ float sW[NOUT][KP];

    const int tid   = threadIdx.x;
    const int lane  = tid & 31;
    const int wave  = tid >> 5;
    const int half  = lane >> 4;
    const int lr    = lane & 15;
    const int base  = blockIdx.x * (WAVES * MT);

    // stage weights (NOUT x K) cooperatively, row-wise float4
    for (int i = tid; i < NOUT * F4ROW; i += 256) {
        int r  = i / F4ROW;
        int c4 = i % F4ROW;
        reinterpret_cast<float4*>(&sW[r][0])[c4] =
            reinterpret_cast<const float4*>(W + (size_t)r * K)[c4];
    }

    // stage this wave's 16 x K A-tile (clamped reads for tail rows)
    for (int i = lane; i < MT * F4ROW; i += 32) {
        int r  = i / F4ROW;
        int c4 = i % F4ROW;
        int g  = base + wave * MT + r;
        if (g >= nodes) g = 0;
        const float* src;
        if (CONCAT)
            src = (g < split) ? (inA + (size_t)g * K)
                              : (inB + (size_t)(g - split) * K);
        else
            src = inA + (size_t)g * K;
        float4 v = reinterpret_cast<const float4*>(src)[c4];
        if (RELU_IN) {
            v.x = fmaxf(v.x, 0.0f); v.y = fmaxf(v.y, 0.0f);
            v.z = fmaxf(v.z, 0.0f); v.w = fmaxf(v.w, 0.0f);
        }
        reinterpret_cast<float4*>(&sA[wave][r][0])[c4] = v;
    }
    __syncthreads();

    v8f acc[NT] = {};

#pragma unroll
    for (int k = 0; k < K; k += 4) {
        // A frag: lanes 0-15 hold K=k..k+1, lanes 16-31 hold K=k+2..k+3 (M=lr)
        v2f a = *reinterpret_cast<const v2f*>(&sA[wave][lr][k + 2 * half]);
#pragma unroll
        for (int nt = 0; nt < NT; ++nt) {
            // B frag: B[kk][n] = W[n][kk]; lane column n = nt*16 + lr
            v2f b = *reinterpret_cast<const v2f*>(&sW[nt * 16 + lr][k + 2 * half]);
            acc[nt] = __builtin_amdgcn_wmma_f32_16x16x4_f32(
                false, a, false, b, (short)0, acc[nt], false, false);
        }
    }

    // D layout: VGPR v -> M = v + 8*half, N = nt*16 + lr
#pragma unroll
    for (int nt = 0; nt < NT; ++nt) {
        float bj = bias[nt * 16 + lr];
#pragma unroll
        for (int v = 0; v < 8; ++v) {
            int node = base + wave * MT + v + 8 * half;
            if (node < nodes)
                out[(size_t)node * NOUT + nt * 16 + lr] = acc[nt][v] + bj;
        }
    }
}

// ---------------------------------------------------------------------------
// Edge scatter:  agg[rows[e], :] += vals[e] * h[cols[e], :]
//   one thread = one (edge, float4-group); L2-resident gather + f32 atomics
// ---------------------------------------------------------------------------
template <int F>
__global__ __launch_bounds__(256) void scatter_kernel(
    const int* __restrict__ rows, const int* __restrict__ cols,
    const float* __restrict__ vals, const float* __restrict__ h,
    float* __restrict__ agg, int nEdges)
{
    constexpr int G = F / 4;
    unsigned t = blockIdx.x * 256u + threadIdx.x;
    int e = (int)(t / G);
    int g = (int)(t % G);
    if (e >= nEdges) return;

    int   r = rows[e];
    int   c = cols[e];
    float v = vals[e];

    float4 hv = *reinterpret_cast<const float4*>(h + (size_t)c * F + g * 4);
    float* dst = agg + (size_t)r * F + g * 4;
    atomicAdd(dst + 0, v * hv.x);
    atomicAdd(dst + 1, v * hv.y);
    atomicAdd(dst + 2, v * hv.z);
    atomicAdd(dst + 3, v * hv.w);
}

// ---------------------------------------------------------------------------
// MLP head: z = relu([bu|bi] @ p1_w.T + p1_b); score = sigmoid(z @ p2_w.T + p2_b)
//   one wave per batch row; lane j computes z_j; shuffle reduction for score
// ---------------------------------------------------------------------------
__global__ __launch_bounds__(256) void mlp_kernel(
    const int* __restrict__ user_ids, const int* __restrict__ item_ids,
    const float* __restrict__ emb,          // [NUM_NODES][EMB], post-ReLU
    const float* __restrict__ p1_w,         // [HID][2*EMB]
    const float* __restrict__ p1_b,         // [HID]
    const float* __restrict__ p2_w,         // [1][HID]
    const float* __restrict__ p2_b,         // [1]
    float* __restrict__ scores)
{
    constexpr int KD = 2 * EMB;             // 128
    __shared__ float sP[HID][KD + 4];       // padded rows
    __shared__ float sZ[8][KD];

    const int tid  = threadIdx.x;
    const int lane = tid & 31;
    const int wave = tid >> 5;
    const int row  = blockIdx.x * 8 + wave;

    // stage p1_w once per block
    for (int i = tid; i < HID * (KD / 4); i += 256) {
        int r  = i / (KD / 4);
        int c4 = i % (KD / 4);
        reinterpret_cast<float4*>(&sP[r][0])[c4] =
            reinterpret_cast<const float4*>(p1_w + (size_t)r * KD)[c4];
    }

    // stage [bu | bi] for this wave's row: lane loads one float4
    int u  = user_ids[row];
    int it = item_ids[row];
    const float* src = (lane < 16)
        ? (emb + (size_t)u * EMB + lane * 4)
        : (emb + (size_t)(NUM_USERS + it) * EMB + (lane - 16) * 4);
    *reinterpret_cast<float4*>(&sZ[wave][lane * 4]) =
        *reinterpret_cast<const float4*>(src);
    __syncthreads();

    // lane j computes z_j (HID == warpSize == 32)
    float z = p1_b[lane];
#pragma unroll 4
    for (int k = 0; k < KD; k += 4) {
        float4 w4 = *reinterpret_cast<const float4*>(&sP[lane][k]);
        z += sZ[wave][k + 0] * w4.x + sZ[wave][k + 1] * w4.y +
             sZ[wave][k + 2] * w4.z + sZ[wave][k + 3] * w4.w;
    }
    z = fmaxf(z, 0.0f);

    float partial = z * p2_w[lane];
#pragma unroll
    for (int off = 16; off > 0; off >>= 1)
        partial += __shfl_down(partial, off, 32);

    if (lane == 0)
        scores[row] = 1.0f / (1.0f + expf(-(partial + p2_b[0])));
}

// ---------------------------------------------------------------------------
// Launch
// ---------------------------------------------------------------------------
extern "C" void kernel_launch(void* const* d_in, const int* in_sizes, int n_in,
                              void* d_out, int out_size, void* d_ws, size_t ws_size,
                              hipStream_t stream) {
    const int*   user_ids   = (const int*)  d_in[0];
    const int*   item_ids   = (const int*)  d_in[1];
    const int*   adj_rows   = (const int*)  d_in[2];
    const int*   adj_cols   = (const int*)  d_in[3];
    const float* adj_vals   = (const float*)d_in[4];
    const float* user_emb_w = (const float*)d_in[5];
    const float* item_emb_w = (const float*)d_in[6];
    const float* gc1_w      = (const float*)d_in[7];
    const float* gc1_b      = (const float*)d_in[8];
    const float* gc2_w      = (const float*)d_in[9];
    const float* gc2_b      = (const float*)d_in[10];
    const float* p1_w       = (const float*)d_in[11];
    const float* p1_b       = (const float*)d_in[12];
    const float* p2_w       = (const float*)d_in[13];
    const float* p2_b       = (const float*)d_in[14];

    float* scores  = (float*)d_out;
    float* emb_out = scores + BATCH;                 // [NUM_NODES][EMB] (users then items)

    float* h1_pre = (float*)d_ws;                    // NUM_NODES * HID
    float* agg1   = h1_pre + (size_t)NUM_NODES * HID;
    float* h2_pre = agg1   + (size_t)NUM_NODES * HID; // NUM_NODES * EMB

    const int nAgg1 = NUM_NODES * HID;               // 3.2M
    const int nEmb  = NUM_NODES * EMB;               // 6.4M
    const int gemmBlocks = (NUM_NODES + 127) / 128;  // 782

    zero_kernel<<<(nAgg1 + 255) / 256, 256, 0, stream>>>(agg1, nAgg1);
    zero_kernel<<<(nEmb  + 255) / 256, 256, 0, stream>>>(emb_out, nEmb);

    // layer 1 linear:  h1_pre = [user_emb;item_emb] @ gc1_w.T + gc1_b
    gc_gemm<EMB, HID, false, true><<<gemmBlocks, 256, 0, stream>>>(
        user_emb_w, item_emb_w, NUM_USERS, gc1_w, gc1_b, h1_pre, NUM_NODES);

    // layer 1 aggregate: agg1[r] += v * h1_pre[c]
    scatter_kernel<HID><<<(N_EDGES * (HID / 4) + 255) / 256, 256, 0, stream>>>(
        adj_rows, adj_cols, adj_vals, h1_pre, agg1, N_EDGES);

    // layer 2 linear (fused relu on input): h2_pre = relu(agg1) @ gc2_w.T + gc2_b
    gc_gemm<HID, EMB, true, false><<<gemmBlocks, 256, 0, stream>>>(
        agg1, nullptr, 0, gc2_w, gc2_b, h2_pre, NUM_NODES);

    // layer 2 aggregate straight into output embedding region
    scatter_kernel<EMB><<<(N_EDGES * (EMB / 4) + 255) / 256, 256, 0, stream>>>(
        adj_rows, adj_cols, adj_vals, h2_pre, emb_out, N_EDGES);

    // final relu (outputs are post-relu embeddings)
    relu_kernel<<<(nEmb + 255) / 256, 256, 0, stream>>>(emb_out, nEmb);

    // MLP ranking head
    mlp_kernel<<<BATCH / 8, 256, 0, stream>>>(
        user_ids, item_ids, emb_out, p1_w, p1_b, p2_w, p2_b, scores);
}